// DPFMLoss_16716012716625
// MI455X (gfx1250) — compile-verified
//
#include <hip/hip_runtime.h>
#include <hip/hip_bf16.h>
#include <math.h>

typedef __attribute__((ext_vector_type(2))) float v2f;
typedef __attribute__((ext_vector_type(8))) float v8f;
typedef __attribute__((ext_vector_type(4))) int   v4i;

#define ACC_FMAP  0
#define ACC_BCE12 1
#define ACC_BCE21 2
#define ACC_NCE   3
#define ACC_GT12  4
#define ACC_GT21  5

#if defined(__has_builtin)
#  if __has_builtin(__builtin_amdgcn_global_load_async_to_lds_b128)
#    define HAVE_ASYNC_LDS 1
#  else
#    define HAVE_ASYNC_LDS 0
#  endif
#  if __has_builtin(__builtin_amdgcn_s_wait_asynccnt)
#    define WAIT_ASYNC(n) __builtin_amdgcn_s_wait_asynccnt(n)
#  else
#    define WAIT_ASYNC(n) asm volatile("s_wait_asynccnt %0" :: "n"(n) : "memory")
#  endif
#else
#  define HAVE_ASYNC_LDS 0
#  define WAIT_ASYNC(n) asm volatile("s_wait_asynccnt %0" :: "n"(n) : "memory")
#endif

// ---------------------------------------------------------------------------
// 1) Frobenius: sum((C12 - C_gt)^2) over K*K elements (batch==1 -> mean==sum)
// ---------------------------------------------------------------------------
__global__ void fmap_kernel(const float* __restrict__ a, const float* __restrict__ b,
                            float* __restrict__ accum, int n) {
  float s = 0.0f;
  for (int i = blockIdx.x * blockDim.x + threadIdx.x; i < n; i += gridDim.x * blockDim.x) {
    float d = a[i] - b[i];
    s += d * d;
  }
  #pragma unroll
  for (int off = 16; off >= 1; off >>= 1) s += __shfl_xor(s, off);
  __shared__ float red[8];
  if ((threadIdx.x & 31) == 0) red[threadIdx.x >> 5] = s;
  __syncthreads();
  if (threadIdx.x == 0) {
    float t = 0.0f;
    for (unsigned w = 0; w < (blockDim.x >> 5); ++w) t += red[w];
    atomicAdd(&accum[ACC_FMAP], t);
  }
}

// ---------------------------------------------------------------------------
// 2) Sum of gt masks (for w_neg = mean(gt))
// ---------------------------------------------------------------------------
__global__ void gtsum_kernel(const int* __restrict__ g12, const int* __restrict__ g21,
                             float* __restrict__ accum, int n) {
  float s12 = 0.0f, s21 = 0.0f;
  for (int i = blockIdx.x * blockDim.x + threadIdx.x; i < n; i += gridDim.x * blockDim.x) {
    s12 += (float)g12[i];
    s21 += (float)g21[i];
  }
  #pragma unroll
  for (int off = 16; off >= 1; off >>= 1) {
    s12 += __shfl_xor(s12, off);
    s21 += __shfl_xor(s21, off);
  }
  __shared__ float r12[8], r21[8];
  if ((threadIdx.x & 31) == 0) { r12[threadIdx.x >> 5] = s12; r21[threadIdx.x >> 5] = s21; }
  __syncthreads();
  if (threadIdx.x == 0) {
    float t12 = 0.0f, t21 = 0.0f;
    for (unsigned w = 0; w < (blockDim.x >> 5); ++w) { t12 += r12[w]; t21 += r21[w]; }
    atomicAdd(&accum[ACC_GT12], t12);
    atomicAdd(&accum[ACC_GT21], t21);
  }
}

// ---------------------------------------------------------------------------
// 3) Weighted BCE (both directions), uses gt sums from previous kernel
// ---------------------------------------------------------------------------
__global__ void bce_kernel(const float* __restrict__ p12, const int* __restrict__ g12,
                           const float* __restrict__ p21, const int* __restrict__ g21,
                           float* __restrict__ accum, int n) {
  const float inv_n = 1.0f / (float)n;
  const float wneg12 = accum[ACC_GT12] * inv_n;
  const float wpos12 = 1.0f - wneg12;
  const float wneg21 = accum[ACC_GT21] * inv_n;
  const float wpos21 = 1.0f - wneg21;
  float s12 = 0.0f, s21 = 0.0f;
  for (int i = blockIdx.x * blockDim.x + threadIdx.x; i < n; i += gridDim.x * blockDim.x) {
    {
      float p = p12[i];
      float gt = (float)g12[i];
      float lp  = fmaxf(__logf(p), -100.0f);
      float l1p = fmaxf(log1pf(-p), -100.0f);
      float bce = -(gt * lp + (1.0f - gt) * l1p);
      float w = (gt >= 0.5f) ? wpos12 : wneg12;
      s12 += w * bce;
    }
    {
      float p = p21[i];
      float gt = (float)g21[i];
      float lp  = fmaxf(__logf(p), -100.0f);
      float l1p = fmaxf(log1pf(-p), -100.0f);
      float bce = -(gt * lp + (1.0f - gt) * l1p);
      float w = (gt >= 0.5f) ? wpos21 : wneg21;
      s21 += w * bce;
    }
  }
  #pragma unroll
  for (int off = 16; off >= 1; off >>= 1) {
    s12 += __shfl_xor(s12, off);
    s21 += __shfl_xor(s21, off);
  }
  __shared__ float r12[8], r21[8];
  if ((threadIdx.x & 31) == 0) { r12[threadIdx.x >> 5] = s12; r21[threadIdx.x >> 5] = s21; }
  __syncthreads();
  if (threadIdx.x == 0) {
    float t12 = 0.0f, t21 = 0.0f;
    for (unsigned w = 0; w < (blockDim.x >> 5); ++w) { t12 += r12[w]; t21 += r21[w]; }
    atomicAdd(&accum[ACC_BCE12], t12);
    atomicAdd(&accum[ACC_BCE21], t21);
  }
}

// ---------------------------------------------------------------------------
// 4) Gather + L2-normalize q = normalize(feat1[map[:,0]]), k = normalize(feat2[map[:,1]])
//    Also store qq[p]=sum(q*q), kk[p]=sum(k*k) (post-normalization, matching ref).
// ---------------------------------------------------------------------------
__global__ __launch_bounds__(128) void gather_norm_kernel(
    const float* __restrict__ feat1, const float* __restrict__ feat2,
    const int* __restrict__ map21,
    float* __restrict__ q, float* __restrict__ kmat,
    float* __restrict__ qq, float* __restrict__ kk) {
  __shared__ float red[4];
  const int r = blockIdx.x;
  const int tid = threadIdx.x;
  const int i0 = map21[2 * r + 0];
  const int i1 = map21[2 * r + 1];
  {
    float x = feat1[(size_t)i0 * 128 + tid];
    float ss = x * x;
    #pragma unroll
    for (int off = 16; off >= 1; off >>= 1) ss += __shfl_xor(ss, off);
    if ((tid & 31) == 0) red[tid >> 5] = ss;
    __syncthreads();
    float tot = red[0] + red[1] + red[2] + red[3];
    float scale = 1.0f / fmaxf(sqrtf(tot), 1e-12f);
    q[(size_t)r * 128 + tid] = x * scale;
    if (tid == 0) qq[r] = tot * scale * scale;
    __syncthreads();
  }
  {
    float x = feat2[(size_t)i1 * 128 + tid];
    float ss = x * x;
    #pragma unroll
    for (int off = 16; off >= 1; off >>= 1) ss += __shfl_xor(ss, off);
    if ((tid & 31) == 0) red[tid >> 5] = ss;
    __syncthreads();
    float tot = red[0] + red[1] + red[2] + red[3];
    float scale = 1.0f / fmaxf(sqrtf(tot), 1e-12f);
    kmat[(size_t)r * 128 + tid] = x * scale;
    if (tid == 0) kk[r] = tot * scale * scale;
  }
}

// ---------------------------------------------------------------------------
// 5) NCE: per 16-row block, stream all 16-col tiles via WMMA f32 16x16x4,
//    online log-softmax per row, diagonal extraction.
//    4 waves split column tiles mod 4. k-tiles are staged with coalesced
//    GLOBAL_LOAD_ASYNC_TO_LDS_B128 (double-buffered per wave, ASYNCcnt-
//    synchronized, no barriers), then read as B-fragments with conflict-free
//    ds_load_b64 (row stride 132 words -> all 64 banks hit exactly once).
// ---------------------------------------------------------------------------
#define TILE_W 132              // 128 floats + 4-word pad (bank-conflict-free)

__global__ __launch_bounds__(128) void nce_kernel(
    const float* __restrict__ q, const float* __restrict__ kmat,
    const float* __restrict__ qq, const float* __restrict__ kk,
    float* __restrict__ accum, int P) {
  const int lane = threadIdx.x & 31;
  const int wave = threadIdx.x >> 5;     // 0..3
  const int col0 = lane & 15;            // N position within tile (and M for A)
  const int half = lane >> 4;            // 0: K{0,1}/rows 0-7, 1: K{2,3}/rows 8-15
  const int rb = blockIdx.x * 16;        // row base of this block
  const int ntiles = P >> 4;
  const float invT = 1.0f / 0.07f;

  __shared__ float sm_m[64], sm_s[64], sm_diag[4];
#if HAVE_ASYNC_LDS
  __shared__ float smem[4 * 2 * 16 * TILE_W];   // 4 waves x 2 buffers x tile
#endif

  // A fragments: row (rb+col0), K-step s covers K = 4s + 2*half + {0,1}
  v2f a[32];
  {
    const float* qrow = q + (size_t)(rb + col0) * 128 + 2 * half;
    #pragma unroll
    for (int s = 0; s < 32; ++s) a[s] = *(const v2f*)(qrow + 4 * s);
  }
  float qqr[8];
  #pragma unroll
  for (int v = 0; v < 8; ++v) qqr[v] = qq[rb + v + 8 * half];

  float m[8], ssum[8];
  #pragma unroll
  for (int v = 0; v < 8; ++v) { m[v] = -__builtin_inff(); ssum[v] = 0.0f; }
  float diag = 0.0f;

#if HAVE_ASYNC_LDS
  float* buf0 = &smem[(wave * 2 + 0) * 16 * TILE_W];
  float* buf1 = &smem[(wave * 2 + 1) * 16 * TILE_W];
  // Stage tile t (16 rows x 512B) with 16 coalesced async B128 copies.
  auto stage = [&](int t, float* buf) {
    const float* g = kmat + (size_t)(t * 16) * 128 + lane * 4;   // 16B per lane
    #pragma unroll
    for (int i = 0; i < 16; ++i) {
      __builtin_amdgcn_global_load_async_to_lds_b128(
          (v4i*)(g + i * 128),
          (v4i*)(buf + i * TILE_W + lane * 4),
          0, 0);
    }
  };

  const int nmy = (ntiles - wave + 3) >> 2;       // tiles handled by this wave
  if (nmy > 0) stage(wave, buf0);
  for (int j = 0; j < nmy; ++j) {
    const int t = wave + 4 * j;
    float* cur = (j & 1) ? buf1 : buf0;
    if (j + 1 < nmy) {
      stage(t + 4, (j & 1) ? buf0 : buf1);
      WAIT_ASYNC(16);   // async loads complete in order: tile t resident
    } else {
      WAIT_ASYNC(0);
    }
    asm volatile("" ::: "memory");
    const float* brow = cur + col0 * TILE_W + 2 * half;
    v8f acc = {};
    #pragma unroll
    for (int s = 0; s < 32; ++s) {
      v2f b = *(const v2f*)(brow + 4 * s);   // ds_load_b64, conflict-free
      acc = __builtin_amdgcn_wmma_f32_16x16x4_f32(
          false, a[s], false, b, (short)0, acc, false, false);
    }
#else
  for (int t = wave; t < ntiles; t += 4) {
    const float* krow = kmat + (size_t)(t * 16 + col0) * 128 + 2 * half;
    v8f acc = {};
    #pragma unroll
    for (int s = 0; s < 32; ++s) {
      v2f b = *(const v2f*)(krow + 4 * s);
      acc = __builtin_amdgcn_wmma_f32_16x16x4_f32(
          false, a[s], false, b, (short)0, acc, false, false);
    }
#endif
    const float kkc = kk[t * 16 + col0];
    const bool diag_tile = (t == (int)blockIdx.x);
    #pragma unroll
    for (int v = 0; v < 8; ++v) {
      float d2 = qqr[v] + kkc - 2.0f * acc[v];
      float logit = -sqrtf(fmaxf(d2, 0.0f)) * invT;
      float mn = fmaxf(m[v], logit);
      ssum[v] = ssum[v] * __expf(m[v] - mn) + __expf(logit - mn);
      m[v] = mn;
      if (diag_tile) {
        // (row,col)=(v+8*half, col0): diagonal held by lanes 0..7 (v==lane)
        // and lanes 24..31 (v==lane-24)
        if ((lane < 8 && v == lane) || (lane >= 24 && v == lane - 24)) diag = logit;
      }
    }
  }

  // Merge (m,s) across the 16-lane group (columns of each row live in one half)
  #pragma unroll
  for (int off = 1; off <= 8; off <<= 1) {
    #pragma unroll
    for (int v = 0; v < 8; ++v) {
      float mo = __shfl_xor(m[v], off);
      float so = __shfl_xor(ssum[v], off);
      float mn = fmaxf(m[v], mo);
      ssum[v] = ssum[v] * __expf(m[v] - mn) + so * __expf(mo - mn);
      m[v] = mn;
    }
  }
  // Per-wave diag partial (sum over lanes; at most one diag per lane)
  float dsum = diag;
  #pragma unroll
  for (int off = 16; off >= 1; off >>= 1) dsum += __shfl_xor(dsum, off);
  if (lane == 0) sm_diag[wave] = dsum;
  // Publish per-row (m,s): lanes 0 and 16 hold rows v+8*half
  if (col0 == 0) {
    #pragma unroll
    for (int v = 0; v < 8; ++v) {
      sm_m[wave * 16 + v + 8 * half] = m[v];
      sm_s[wave * 16 + v + 8 * half] = ssum[v];
    }
  }
  __syncthreads();

  // Wave 0, lanes 0..15: merge the 4 waves per row, lse = m + log(s)
  if (threadIdx.x < 16) {
    int row = threadIdx.x;
    float mm = sm_m[row], ss = sm_s[row];
    #pragma unroll
    for (int w = 1; w < 4; ++w) {
      float mo = sm_m[w * 16 + row];
      float so = sm_s[w * 16 + row];
      float mn = fmaxf(mm, mo);
      ss = ss * __expf(mm - mn) + so * __expf(mo - mn);
      mm = mn;
    }
    float part = mm + __logf(ss);            // logsumexp of this row
    #pragma unroll
    for (int off = 8; off >= 1; off >>= 1) part += __shfl_xor(part, off);
    if (row == 0) {
      part -= (sm_diag[0] + sm_diag[1] + sm_diag[2] + sm_diag[3]);
      atomicAdd(&accum[ACC_NCE], part);
    }
  }
}

// ---------------------------------------------------------------------------
// 6) Final combine: out = fmap*W_FMAP + (bce12+bce21)/N*W_ACC + nce/P*W_NCE
// ---------------------------------------------------------------------------
__global__ void combine_kernel(float* __restrict__ out, const float* __restrict__ accum,
                               int n, int p) {
  if (threadIdx.x == 0 && blockIdx.x == 0) {
    const float W_FMAP = 1.0f, W_ACC = 1.0f, W_NCE = 0.1f;
    float fmap = accum[ACC_FMAP];
    float acc_l = (accum[ACC_BCE12] + accum[ACC_BCE21]) / (float)n;
    float nce = accum[ACC_NCE] / (float)p;
    out[0] = fmap * W_FMAP + acc_l * W_ACC + nce * W_NCE;
  }
}

extern "C" void kernel_launch(void* const* d_in, const int* in_sizes, int n_in,
                              void* d_out, int out_size, void* d_ws, size_t ws_size,
                              hipStream_t stream) {
  const float* C12   = (const float*)d_in[0];
  const float* C_gt  = (const float*)d_in[1];
  const int*   map21 = (const int*)d_in[2];
  const float* feat1 = (const float*)d_in[3];
  const float* feat2 = (const float*)d_in[4];
  const float* ov12  = (const float*)d_in[5];
  const float* ov21  = (const float*)d_in[6];
  const int*   g12   = (const int*)d_in[7];
  const int*   g21   = (const int*)d_in[8];

  const int KK = in_sizes[0];        // K*K (batch 1)
  const int P  = in_sizes[2] / 2;    // 4096
  const int N  = in_sizes[5];        // 100000
  const int D  = 128;                // in_sizes[3] / N

  float* ws    = (float*)d_ws;
  float* accum = ws;                                  // 16 floats
  float* q     = ws + 16;                             // P*D
  float* kbuf  = q + (size_t)P * D;                   // P*D
  float* qq    = kbuf + (size_t)P * D;                // P
  float* kk    = qq + P;                              // P

  (void)hipMemsetAsync(accum, 0, 16 * sizeof(float), stream);

  fmap_kernel<<<40, 256, 0, stream>>>(C12, C_gt, accum, KK);
  gtsum_kernel<<<256, 256, 0, stream>>>(g12, g21, accum, N);
  gather_norm_kernel<<<P, 128, 0, stream>>>(feat1, feat2, map21, q, kbuf, qq, kk);
  bce_kernel<<<256, 256, 0, stream>>>(ov12, g12, ov21, g21, accum, N);
  nce_kernel<<<P / 16, 128, 0, stream>>>(q, kbuf, qq, kk, accum, P);
  combine_kernel<<<1, 32, 0, stream>>>((float*)d_out, accum, N, P);
}